// CedrDrmmRanker_6305011990556
// MI455X (gfx1250) — compile-verified
//
#include <hip/hip_runtime.h>
#include <math.h>

typedef __attribute__((ext_vector_type(2))) float v2f;
typedef __attribute__((ext_vector_type(4))) float f4;
typedef __attribute__((ext_vector_type(8))) float v8f;

#define H    768
#define S    512
#define QL   20
#define ND   492
#define NB   16
#define NL   14
#define QSTR 772   // 768+4: rows 16B aligned, 4*row mod 64 distinct -> conflict-free A reads
#define KC   32    // K chunk
#define NCH  (H/KC)        // 24 chunks
#define DSTR 36    // 32+4: 36*n mod 64 distinct for n=0..15 -> conflict-free B reads; 144B rows (16B aligned)
#define DBUF (128*DSTR)    // one D buffer, floats
#define NT   128   // N tile (8 waves * 16)

// dynamic LDS layout (floats):
//   smQ   [32*QSTR]  = 24704
//   smD   [2*DBUF]   =  9216   (double-buffered async staging)
//   smQn  [32]
//   smDn  [128]
//   smHist[16] (as unsigned)
#define SMEM_FLOATS (32*QSTR + 2*DBUF + 32 + 128 + 16)

__global__ void __launch_bounds__(256)
cedr_hist_kernel(const float* __restrict__ hs,
                 const float* __restrict__ Whist,
                 const float* __restrict__ bhist,
                 const float* __restrict__ Wcomb,
                 float* __restrict__ ws)
{
    extern __shared__ float smem[];
    float*    smQ    = smem;
    float*    smD    = smQ + 32*QSTR;
    float*    smQn   = smD + 2*DBUF;
    float*    smDn   = smQn + 32;
    unsigned* smHist = (unsigned*)(smDn + 128);

    const int tid = threadIdx.x;
    const int l   = blockIdx.x >> 4;       // 0..13 (all_layers index)
    const int b   = blockIdx.x & 15;
    const int hl  = (l == 0) ? 0 : (l - 1);
    const float* base = hs + ((size_t)(hl * NB + b)) * S * H;

    // ---- stage Q: rows 0..19 real, 20..31 zero-padded ----
    for (int i = tid; i < 32 * (H/4); i += 256) {
        int row = i / (H/4);
        int c4  = i % (H/4);
        f4 v = {0.f, 0.f, 0.f, 0.f};
        if (row < QL) v = *(const f4*)(base + (size_t)row * H + c4*4);
        *(f4*)(smQ + row*QSTR + c4*4) = v;
    }
    if (tid < 11) smHist[tid] = 0u;
    __syncthreads();

    // ---- q norms (deterministic, tiny) ----
    if (tid < 32) {
        float s = 0.f;
        if (tid < QL) {
            const float* r = smQ + tid*QSTR;
            for (int k = 0; k < H; ++k) s += r[k]*r[k];
        }
        smQn[tid] = sqrtf(s);
    }

    const int wave   = tid >> 5;
    const int lane   = tid & 31;
    const int laneN  = lane & 15;
    const int laneHi = lane >> 4;

    // async staging: 128 rows x 32 floats per chunk; 8 threads/row, b128 each, 4 rows/thread
    const int rgrp = tid >> 3;             // 0..31
    const int cseg = tid & 7;              // 16B column segment

    for (int nt = 0; nt < 4; ++nt) {
        const int nbase = nt * NT;
        if (tid < 128) smDn[tid] = 0.f;    // prior hist consumed behind trailing barrier
        v8f C0 = {};
        v8f C1 = {};

        // issue_chunk(kc, buf): per-thread 4 async b128 loads, OOB rows skipped (lanes masked)
        auto issue_chunk = [&](int kc, int bufIdx) {
            float* bufBase = smD + bufIdx * DBUF;
            #pragma unroll
            for (int p = 0; p < 4; ++p) {
                int row = p*32 + rgrp;
                int gn  = nbase + row;
                if (gn < ND) {
                    unsigned ldsOff = (unsigned)(uintptr_t)(bufBase + row*DSTR + cseg*4);
                    const float* g = base + (size_t)(QL + gn) * H + kc*KC + cseg*4;
                    asm volatile("global_load_async_to_lds_b128 %0, %1, off"
                                 :: "v"(ldsOff), "v"(g) : "memory");
                }
            }
        };

        issue_chunk(0, 0);

        for (int kc = 0; kc < NCH; ++kc) {
            // own chunk-kc loads done; barrier => everyone's done + prior compute finished
            asm volatile("s_wait_asynccnt 0x0" ::: "memory");
            __syncthreads();

            if (kc + 1 < NCH) issue_chunk(kc + 1, (kc + 1) & 1);   // overlaps with compute below

            const float* buf = smD + (kc & 1) * DBUF;

            // deterministic per-row sum-of-squares for d norms
            if (tid < 128) {
                const float* r = buf + tid*DSTR;
                float s = 0.f;
                for (int k = 0; k < KC; ++k) s += r[k]*r[k];
                smDn[tid] += s;
            }

            // WMMA over this K chunk: A from smQ (full K resident), B from current D buffer
            const float* qa0 = smQ + laneN*QSTR       + kc*KC + laneHi*2;
            const float* qa1 = smQ + (16+laneN)*QSTR  + kc*KC + laneHi*2;
            const float* bb  = buf + (wave*16+laneN)*DSTR     + laneHi*2;
            #pragma unroll
            for (int ks = 0; ks < KC; ks += 4) {
                v2f a0 = *(const v2f*)(qa0 + ks);      // K = ks+{0,1} / {2,3} per half-wave
                v2f a1 = *(const v2f*)(qa1 + ks);
                v2f bf = *(const v2f*)(bb  + ks);
                C0 = __builtin_amdgcn_wmma_f32_16x16x4_f32(
                        false, a0, false, bf, (short)0, C0, false, false);
                C1 = __builtin_amdgcn_wmma_f32_16x16x4_f32(
                        false, a1, false, bf, (short)0, C1, false, false);
            }
        }
        __syncthreads();

        // ---- cosine normalize + histogram (integer LDS atomics: deterministic) ----
        const int n = nbase + wave*16 + laneN;
        if (n < ND) {
            float dn = sqrtf(smDn[wave*16 + laneN]);
            #pragma unroll
            for (int r = 0; r < 8; ++r) {
                int m0 = r + laneHi*8;                  // C layout: lanes 16-31 hold M=r+8
                if (m0 < QL) {
                    float denom = fmaxf(smQn[m0] * dn, 1e-8f);
                    float sim = C0[r] / denom;
                    int bi = (int)floorf((sim + 1.0f) * 5.5f);
                    if (bi >= 0 && bi < 11) atomicAdd(&smHist[bi], 1u);
                }
                int m1 = 16 + m0;                       // rows 16..19 real, rest padded
                if (m1 < QL) {
                    float denom = fmaxf(smQn[m1] * dn, 1e-8f);
                    float sim = C1[r] / denom;
                    int bi = (int)floorf((sim + 1.0f) * 5.5f);
                    if (bi >= 0 && bi < 11) atomicAdd(&smHist[bi], 1u);
                }
            }
        }
        __syncthreads();   // hist/smDn consumed before next tile resets & re-stages
    }

    // ---- fold histogram through W_hist/b_hist and the W_comb slice ----
    if (tid == 0) {
        const float inv = 1.0f / (float)(QL * ND);
        float partial = 0.f;
        for (int o = 0; o < 5; ++o) {
            float hf = bhist[o];
            for (int nb2 = 0; nb2 < 11; ++nb2)
                hf += (float)smHist[nb2] * inv * Whist[o*11 + nb2];
            partial += hf * Wcomb[H + l*5 + o];
        }
        ws[l*NB + b] = partial;
    }
}

__global__ void __launch_bounds__(256)
cedr_finalize_kernel(const float* __restrict__ hs,
                     const float* __restrict__ Wcomb,
                     const float* __restrict__ bcomb,
                     const float* __restrict__ part,
                     float* __restrict__ out)
{
    __shared__ float red[256];
    const int b = blockIdx.x;              // 0..15
    // cls_output = hidden_states[-1][:, 0, :]  -> layer index 12, s = 0
    const float* cls = hs + ((size_t)(12 * NB + b)) * S * H;
    float s = 0.f;
    for (int h = threadIdx.x; h < H; h += 256) s += cls[h] * Wcomb[h];
    red[threadIdx.x] = s;
    __syncthreads();
    for (int off = 128; off > 0; off >>= 1) {
        if (threadIdx.x < off) red[threadIdx.x] += red[threadIdx.x + off];
        __syncthreads();
    }
    if (threadIdx.x == 0) {
        float acc = red[0] + bcomb[0];
        for (int ll = 0; ll < NL; ++ll) acc += part[ll*NB + b];
        out[b] = acc;
    }
}

extern "C" void kernel_launch(void* const* d_in, const int* in_sizes, int n_in,
                              void* d_out, int out_size, void* d_ws, size_t ws_size,
                              hipStream_t stream) {
    const float* hs    = (const float*)d_in[0];   // (13,16,512,768) f32
    const float* Whist = (const float*)d_in[1];   // (5,11)
    const float* bhist = (const float*)d_in[2];   // (5,)
    const float* Wcomb = (const float*)d_in[3];   // (1,838)
    const float* bcomb = (const float*)d_in[4];   // (1,)
    float*       out   = (float*)d_out;           // (16,1)
    float*       ws    = (float*)d_ws;            // 14*16 partials

    const size_t shmem = (size_t)SMEM_FLOATS * sizeof(float);   // ~136 KB (LDS is 320 KB/WGP)
    cedr_hist_kernel<<<dim3(NL * NB), dim3(256), shmem, stream>>>(hs, Whist, bhist, Wcomb, ws);
    cedr_finalize_kernel<<<dim3(NB), dim3(256), 0, stream>>>(hs, Wcomb, bcomb, ws, out);
}